// InteractionBlock_65962107732486
// MI455X (gfx1250) — compile-verified
//
#include <hip/hip_runtime.h>
#include <hip/hip_bf16.h>

// ---------------------------------------------------------------------------
// GemNet InteractionBlock forward for MI455X (gfx1250), fp32 throughout.
// All dense contractions use V_WMMA_F32_16X16X4_F32 (wave32 WMMA).
// ---------------------------------------------------------------------------

#define INV_SQRT2f 0.70710678118654752440f

typedef __attribute__((ext_vector_type(2))) float v2f;
typedef __attribute__((ext_vector_type(8))) float v8f;

__device__ __forceinline__ float silu_f(float x) {
  return x * (1.0f / (1.0f + __expf(-x)));
}

__device__ __forceinline__ v8f wmma4(v2f a, v2f b, v8f c) {
  // D = A(16x4,f32) x B(4x16,f32) + C(16x16,f32)
  return __builtin_amdgcn_wmma_f32_16x16x4_f32(
      /*neg_a=*/false, a, /*neg_b=*/false, b,
      /*c_mod=*/(short)0, c, /*reuse_a=*/false, /*reuse_b=*/false);
}

__device__ __forceinline__ float4 ld4(const float* p) {
  return *(const float4*)p;
}

enum { EPI_NONE = 0, EPI_SILU = 1, EPI_SILU_SCALE = 2, EPI_SILU_RES = 3 };

// ---------------------------------------------------------------------------
// Generic WMMA GEMM:  C(M x Nout) = epi( A(M x K) @ Bw(K x Nout) )
// block = 128 threads (4 waves). Each wave: 16 rows x 32 cols (2 accumulators).
// Workgroup tile: 64 rows x 32 cols. grid = (ceil(M/64), Nout/32).
// A tile staged in LDS (row stride K+4 -> bank-conflict free WMMA reads).
// CONCAT mode: rows are [ A[gS[r]]:128 | A[gT[r]]:128 | Xc[r]:128 ] (K==384).
// ---------------------------------------------------------------------------
template <int K, int EPI, bool CONCAT>
__global__ __launch_bounds__(128) void gemm_kernel(
    const float* __restrict__ A, const float* __restrict__ Bw,
    float* __restrict__ C, int M, int Nout,
    const float* __restrict__ Sc, const float* __restrict__ R,
    const int* __restrict__ gS, const int* __restrict__ gT,
    const float* __restrict__ Xc) {
  constexpr int KP = K + 4;
  __shared__ float As[64 * KP];

  const int tid = threadIdx.x;
  const int rowBase = blockIdx.x * 64;
  const int cBase = blockIdx.y * 32;

  // ---- stage A tile (64 x K) into LDS, coalesced float4 ----
  constexpr int ROW4 = K / 4;
  for (int i = tid; i < 64 * ROW4; i += 128) {
    const int row = i / ROW4;
    const int c4 = i % ROW4;
    const int grow = rowBase + row;
    float4 v = {0.f, 0.f, 0.f, 0.f};
    if (grow < M) {
      if constexpr (CONCAT) {
        const int col = c4 * 4;
        if (col < 128)
          v = ld4(&A[(size_t)gS[grow] * 128 + col]);
        else if (col < 256)
          v = ld4(&A[(size_t)gT[grow] * 128 + (col - 128)]);
        else
          v = ld4(&Xc[(size_t)grow * 128 + (col - 256)]);
      } else {
        v = ld4(&A[(size_t)grow * K + c4 * 4]);
      }
    }
    *(float4*)&As[row * KP + c4 * 4] = v;
  }
  __syncthreads();

  const int w = tid >> 5;
  const int lane = tid & 31;
  const int n = lane & 15;    // B column / A row within fragment
  const int hf = lane >> 4;   // lane half selects K pair {0,1} vs {2,3}
  const int r0 = w * 16;

  v8f acc0 = {0.f, 0.f, 0.f, 0.f, 0.f, 0.f, 0.f, 0.f};
  v8f acc1 = {0.f, 0.f, 0.f, 0.f, 0.f, 0.f, 0.f, 0.f};
  const float* Arow = &As[(r0 + n) * KP];

#pragma unroll 4
  for (int k = 0; k < K; k += 4) {
    const int kk = k + 2 * hf;
    v2f a;
    a.x = Arow[kk];
    a.y = Arow[kk + 1];
    const float* b0p = &Bw[(size_t)kk * Nout + cBase + n];
    const float* b1p = &Bw[(size_t)(kk + 1) * Nout + cBase + n];
    v2f b0, b1;
    b0.x = b0p[0];
    b0.y = b1p[0];
    b1.x = b0p[16];
    b1.y = b1p[16];
    acc0 = wmma4(a, b0, acc0);
    acc1 = wmma4(a, b1, acc1);
  }

  // ---- epilogue; C layout: lane n = col, row m = 8*half + v ----
#pragma unroll
  for (int v = 0; v < 8; ++v) {
    const int row = rowBase + r0 + 8 * hf + v;
    if (row >= M) continue;
    const size_t o0 = (size_t)row * Nout + cBase + n;
    const size_t o1 = o0 + 16;
    float y0 = acc0[v], y1 = acc1[v];
    if constexpr (EPI == EPI_SILU) {
      y0 = silu_f(y0);
      y1 = silu_f(y1);
    } else if constexpr (EPI == EPI_SILU_SCALE) {
      y0 = silu_f(y0) * Sc[o0];
      y1 = silu_f(y1) * Sc[o1];
    } else if constexpr (EPI == EPI_SILU_RES) {
      y0 = (R[o0] + silu_f(y0)) * INV_SQRT2f;
      y1 = (R[o1] + silu_f(y1)) * INV_SQRT2f;
    }
    C[o0] = y0;
    C[o1] = y1;
  }
}

// ---------------------------------------------------------------------------
// Fused EfficientInteractionBilinear. One workgroup = 16 edges, 4 waves.
//   m2[e,k,:]   = m_kt[id3_kt[e*8+k], :]            (gather, LDS)
//   sum_k[e,s,c]= sum_k sph[e,k,s] * m2[e,k,c]      (LDS)
//   a[e,j,c]    = sum_s rbfW1[e,j,s] * sum_k[e,s,c] (LDS, flattened p=j*64+c)
//   xb[e,u]     = sum_p a[e,p] * WP[p,u]            (WMMA: M=16,K=1024,N=64)
// WP[p,u] = W_bil[c,j,u] with p=j*64+c (prepacked).
// ---------------------------------------------------------------------------
__global__ __launch_bounds__(128) void bilinear_kernel(
    const float* __restrict__ mkt, const float* __restrict__ sph,
    const float* __restrict__ rbfW1, const float* __restrict__ WP,
    const int* __restrict__ id3_kt, float* __restrict__ xb) {
  constexpr int KP2 = 1024 + 4;
  __shared__ float sSph[16 * 8 * 8];   // [le][k][s pad 8]
  __shared__ float sRbf[16 * 16 * 8];  // [le][j][s pad 8]
  __shared__ float sM2[16 * 8 * 64];   // [le][k][c]
  __shared__ float sSum[16 * 7 * 64];  // [le][s][c]
  __shared__ float sA[16 * KP2];       // [le][p] padded
  __shared__ int sIdx[128];

  const int tid = threadIdx.x;
  const int e0 = blockIdx.x * 16;

  sIdx[tid] = id3_kt[e0 * 8 + tid];  // 16 edges * 8 k = 128 entries

  for (int i = tid; i < 16 * 56; i += 128) {  // sph (E,K=8,S=7)
    const int le = i / 56, rem = i % 56;
    const int k = rem / 7, s = rem % 7;
    sSph[(le * 8 + k) * 8 + s] = sph[(size_t)(e0 + le) * 56 + rem];
  }
  for (int i = tid; i < 16 * 112; i += 128) {  // rbfW1 (E,DC=16,S=7)
    const int le = i / 112, rem = i % 112;
    const int j = rem / 7, s = rem % 7;
    sRbf[(le * 16 + j) * 8 + s] = rbfW1[(size_t)(e0 + le) * 112 + rem];
  }
  __syncthreads();

  for (int i = tid; i < 16 * 8 * 16; i += 128) {  // gather m_kt rows (float4)
    const int le = i >> 7, rem = i & 127;
    const int k = rem >> 4, c4 = rem & 15;
    const int srow = sIdx[le * 8 + k];
    *(float4*)&sM2[(le * 8 + k) * 64 + c4 * 4] =
        ld4(&mkt[(size_t)srow * 64 + c4 * 4]);
  }
  __syncthreads();

  for (int o = tid; o < 16 * 7 * 64; o += 128) {  // sum_k
    const int le = o / 448, rem = o % 448;
    const int s = rem >> 6, c = rem & 63;
    float acc = 0.f;
#pragma unroll
    for (int k = 0; k < 8; ++k)
      acc += sSph[(le * 8 + k) * 8 + s] * sM2[(le * 8 + k) * 64 + c];
    sSum[(le * 7 + s) * 64 + c] = acc;
  }
  __syncthreads();

  for (int o = tid; o < 16 * 16 * 64; o += 128) {  // a[e,j,c] -> sA[e, j*64+c]
    const int le = o >> 10, rem = o & 1023;
    const int j = rem >> 6, c = rem & 63;
    float acc = 0.f;
#pragma unroll
    for (int s = 0; s < 7; ++s)
      acc += sRbf[(le * 16 + j) * 8 + s] * sSum[(le * 7 + s) * 64 + c];
    sA[le * KP2 + j * 64 + c] = acc;
  }
  __syncthreads();

  // ---- WMMA: (16 edges x 1024) @ (1024 x 64); wave w owns 16 columns ----
  const int w = tid >> 5;
  const int lane = tid & 31;
  const int n = lane & 15;
  const int hf = lane >> 4;
  const int cb = w * 16;

  v8f acc = {0.f, 0.f, 0.f, 0.f, 0.f, 0.f, 0.f, 0.f};
  const float* Ar = &sA[n * KP2];
  for (int p = 0; p < 1024; p += 4) {
    const int kk = p + 2 * hf;
    v2f a;
    a.x = Ar[kk];
    a.y = Ar[kk + 1];
    v2f b;
    b.x = WP[(size_t)kk * 64 + cb + n];
    b.y = WP[(size_t)(kk + 1) * 64 + cb + n];
    acc = wmma4(a, b, acc);
  }
#pragma unroll
  for (int v = 0; v < 8; ++v) {
    const int m = 8 * hf + v;
    xb[(size_t)(e0 + m) * 64 + cb + n] = acc[v];
  }
}

// ---------------------------------------------------------------------------
// small elementwise / scatter kernels
// ---------------------------------------------------------------------------
__global__ void wprep_kernel(const float* __restrict__ Wb,
                             float* __restrict__ Wp) {
  // Wp[p*64+u] = W_bil[c,j,u], p = j*64 + c; W_bil shape (64,16,64)
  const int i = blockIdx.x * 256 + threadIdx.x;
  if (i >= 1024 * 64) return;
  const int u = i & 63;
  const int p = i >> 6;
  const int j = p >> 6;
  const int c = p & 63;
  Wp[i] = Wb[(size_t)c * 1024 + j * 64 + u];
}

__global__ void merge_kernel(const float* __restrict__ xskip,
                             const float* __restrict__ xst,
                             const float* __restrict__ xts,
                             const int* __restrict__ swp,
                             float* __restrict__ out, int ne) {
  const int i = blockIdx.x * 256 + threadIdx.x;
  if (i >= ne * 128) return;
  const int e = i >> 7, c = i & 127;
  const float x3 = (xst[i] + xts[(size_t)swp[e] * 128 + c]) * INV_SQRT2f;
  out[i] = (xskip[i] + x3) * INV_SQRT2f;
}

__global__ void addscale_kernel(const float* __restrict__ a,
                                const float* __restrict__ b,
                                float* __restrict__ out, int n) {
  const int i = blockIdx.x * 256 + threadIdx.x;
  if (i < n) out[i] = (a[i] + b[i]) * INV_SQRT2f;
}

__global__ void zero_kernel(float* __restrict__ p, int n) {
  const int i = blockIdx.x * 256 + threadIdx.x;
  if (i < n) p[i] = 0.f;
}

__global__ void scatter_kernel(const float* __restrict__ x,
                               const float* __restrict__ sc,
                               const int* __restrict__ idxt,
                               float* __restrict__ h2, int ne) {
  const int i = blockIdx.x * 256 + threadIdx.x;
  if (i >= ne * 128) return;
  const int e = i >> 7, c = i & 127;
  unsafeAtomicAdd(&h2[(size_t)idxt[e] * 128 + c], x[i] * sc[i]);
}

// ---------------------------------------------------------------------------
// host side
// ---------------------------------------------------------------------------
static void res_layer(hipStream_t s, float* x, float* tmp, int M,
                      const float* W01, float* out) {
  dim3 g((M + 63) / 64, 4);
  // tmp = silu(x @ W0)
  gemm_kernel<128, EPI_SILU, false><<<g, 128, 0, s>>>(
      x, W01, tmp, M, 128, nullptr, nullptr, nullptr, nullptr, nullptr);
  // out = (x + silu(tmp @ W1)) * 1/sqrt(2)
  gemm_kernel<128, EPI_SILU_RES, false><<<g, 128, 0, s>>>(
      tmp, W01 + 128 * 128, out, M, 128, nullptr, x, nullptr, nullptr, nullptr);
}

extern "C" void kernel_launch(void* const* d_in, const int* in_sizes, int n_in,
                              void* d_out, int out_size, void* d_ws,
                              size_t ws_size, hipStream_t stream) {
  (void)in_sizes; (void)n_in; (void)out_size; (void)ws_size;

  constexpr int NA = 10000;   // atoms
  constexpr int E = 120000;   // edges

  const float* m_st = (const float*)d_in[1];
  const float* rbf_h = (const float*)d_in[2];
  const float* rbf3 = (const float*)d_in[3];
  const float* cbf_rbfW1 = (const float*)d_in[4];
  const float* cbf_sph = (const float*)d_in[5];
  const float* W_mlp_st = (const float*)d_in[6];
  const float* W_m_kt = (const float*)d_in[7];
  const float* W_t_rbf = (const float*)d_in[8];
  const float* W_down = (const float*)d_in[9];
  const float* W_bil = (const float*)d_in[10];
  const float* W_st3 = (const float*)d_in[11];
  const float* W_ts3 = (const float*)d_in[12];
  const float* res_before = (const float*)d_in[13];
  const float* res_after = (const float*)d_in[14];
  const float* W_ae_rbf = (const float*)d_in[15];
  const float* W_ae_in = (const float*)d_in[16];
  const float* res_atom = (const float*)d_in[17];
  const float* W_asi = (const float*)d_in[18];
  const float* res_m = (const float*)d_in[19];
  const int* idx_s = (const int*)d_in[20];
  const int* idx_t = (const int*)d_in[21];
  const int* idx_swap = (const int*)d_in[22];
  const int* id3_kt = (const int*)d_in[23];

  // workspace carve-out (~323 MB)
  float* ws = (float*)d_ws;
  size_t off = 0;
  auto alloc = [&](size_t n) { float* p = ws + off; off += n; return p; };
  float* B0 = alloc((size_t)E * 128);   // persistent edge feature x
  float* B1 = alloc((size_t)E * 128);   // x_skip
  float* B2 = alloc((size_t)E * 128);   // temp
  float* B3 = alloc((size_t)E * 128);   // temp
  float* MKT = alloc((size_t)E * 64);   // m_kt (down-projected)
  float* XB = alloc((size_t)E * 64);    // bilinear output
  float* WP = alloc(1024 * 64);         // repacked W_bil
  float* H0 = alloc((size_t)NA * 128);  // h2 accumulation
  float* H1 = alloc((size_t)NA * 128);
  float* H2 = alloc((size_t)NA * 128);

  float* out_h = (float*)d_out;                  // (NA,128)
  float* out_m = out_h + (size_t)NA * 128;       // (E,128)

  const dim3 gE128((E + 63) / 64, 4);
  const dim3 gE64((E + 63) / 64, 2);
  const dim3 gN128((NA + 63) / 64, 4);
  const int nbE = (E * 128 + 255) / 256;

  // 0) repack W_bil
  wprep_kernel<<<(1024 * 64 + 255) / 256, 256, 0, stream>>>(W_bil, WP);
  // 1) B2 = rbf3 @ W_t_rbf   (K=16)
  gemm_kernel<16, EPI_NONE, false><<<gE128, 128, 0, stream>>>(
      rbf3, W_t_rbf, B2, E, 128, nullptr, nullptr, nullptr, nullptr, nullptr);
  // 2) B3 = silu(m_st @ W_m_kt) * B2
  gemm_kernel<128, EPI_SILU_SCALE, false><<<gE128, 128, 0, stream>>>(
      m_st, W_m_kt, B3, E, 128, B2, nullptr, nullptr, nullptr, nullptr);
  // 3) MKT = silu(B3 @ W_down)   (N=64)
  gemm_kernel<128, EPI_SILU, false><<<gE64, 128, 0, stream>>>(
      B3, W_down, MKT, E, 64, nullptr, nullptr, nullptr, nullptr, nullptr);
  // 4) B1 = m_st @ W_mlp_st  (x_skip)
  gemm_kernel<128, EPI_NONE, false><<<gE128, 128, 0, stream>>>(
      m_st, W_mlp_st, B1, E, 128, nullptr, nullptr, nullptr, nullptr, nullptr);
  // 5) fused bilinear -> XB
  bilinear_kernel<<<E / 16, 128, 0, stream>>>(MKT, cbf_sph, cbf_rbfW1, WP,
                                              id3_kt, XB);
  // 6) B2 = silu(XB @ W_st3); B3 = silu(XB @ W_ts3)  (K=64)
  gemm_kernel<64, EPI_SILU, false><<<gE128, 128, 0, stream>>>(
      XB, W_st3, B2, E, 128, nullptr, nullptr, nullptr, nullptr, nullptr);
  gemm_kernel<64, EPI_SILU, false><<<gE128, 128, 0, stream>>>(
      XB, W_ts3, B3, E, 128, nullptr, nullptr, nullptr, nullptr, nullptr);
  // 7) B0 = (B1 + (B2 + B3[idx_swap]) / sqrt2) / sqrt2
  merge_kernel<<<nbE, 256, 0, stream>>>(B1, B2, B3, idx_swap, B0, E);
  // 8) res_before (1 layer)
  res_layer(stream, B0, B2, E, res_before, B0);
  // 9) B0 = (B0 + m_st) / sqrt2
  addscale_kernel<<<nbE, 256, 0, stream>>>(B0, m_st, B0, E * 128);
  // 10) res_after (2 layers)
  res_layer(stream, B0, B2, E, res_after, B0);
  res_layer(stream, B0, B2, E, res_after + 2 * 128 * 128, B0);
  // 11) B2 = rbf_h @ W_ae_rbf
  gemm_kernel<16, EPI_NONE, false><<<gE128, 128, 0, stream>>>(
      rbf_h, W_ae_rbf, B2, E, 128, nullptr, nullptr, nullptr, nullptr, nullptr);
  // 12) H0 = segment_sum(B0 * B2, idx_t)
  zero_kernel<<<(NA * 128 + 255) / 256, 256, 0, stream>>>(H0, NA * 128);
  scatter_kernel<<<nbE, 256, 0, stream>>>(B0, B2, idx_t, H0, E);
  // 13) H1 = silu(H0 @ W_ae_in)
  gemm_kernel<128, EPI_SILU, false><<<gN128, 128, 0, stream>>>(
      H0, W_ae_in, H1, NA, 128, nullptr, nullptr, nullptr, nullptr, nullptr);
  // 14) res_atom (2 layers), final layer writes h_new into d_out
  res_layer(stream, H1, H2, NA, res_atom, H1);
  res_layer(stream, H1, H2, NA, res_atom + 2 * 128 * 128, out_h);
  // 15) B2 = silu(concat[h_new[idx_s], h_new[idx_t], B0] @ W_asi)  (K=384)
  gemm_kernel<384, EPI_SILU, true><<<gE128, 128, 0, stream>>>(
      out_h, W_asi, B2, E, 128, nullptr, nullptr, idx_s, idx_t, B0);
  // 16) res_m (2 layers)
  res_layer(stream, B2, B3, E, res_m, B2);
  res_layer(stream, B2, B3, E, res_m + 2 * 128 * 128, B2);
  // 17) m = (B2 + B0) / sqrt2 -> d_out
  addscale_kernel<<<nbE, 256, 0, stream>>>(B2, B0, out_m, E * 128);
}